// Base_584115552635
// MI455X (gfx1250) — compile-verified
//
#include <hip/hip_runtime.h>
#include <math.h>

typedef __attribute__((ext_vector_type(2))) float v2f;
typedef __attribute__((ext_vector_type(8))) float v8f;

#define NRAYS   262144
#define NS      64          // coarse samples per ray
#define NI      128         // importance samples per ray
#define RPW     16          // rays per wave (WMMA M dim)
#define WAVES   4           // waves per block
#define RPB     (RPW*WAVES) // 64 rays per block
#define STRIDE  68          // padded LDS row stride in floats (16B aligned, conflict-free)

__global__ __launch_bounds__(128) void nerf_resample_wmma(
    const float* __restrict__ nearp, const float* __restrict__ farp,
    const float* __restrict__ ray_o, const float* __restrict__ ray_d,
    const float* __restrict__ alpha, float* __restrict__ out)
{
    // per-wave scratch: bufA holds alpha, later reused for the raw CDF rows
    //                   bufB holds log(1-a+1e-10), later reused for shifted weights
    __shared__ float bufA[WAVES][RPW * STRIDE];
    __shared__ float bufB[WAVES][RPW * STRIDE];

    const int tid   = threadIdx.x;
    const int lane  = tid & 31;
    const int wave  = tid >> 5;
    const int row16 = lane & 15;   // M index (ray within wave-tile) / N index
    const int hh    = lane >> 4;   // lane half selects K pair / row half

    const int blockRay0 = blockIdx.x * RPB;
    const int waveRay0  = blockRay0 + wave * RPW;

    float* A = bufA[wave];
    float* B = bufB[wave];

    // ---------------- Phase 1: coalesced load of alpha[16x64] ----------------
    {
        const float* ga = alpha + (size_t)waveRay0 * NS;
        #pragma unroll
        for (int it = 0; it < 8; ++it) {
            int idx = it * 128 + lane * 4;       // 16B-granular, fully coalesced
            int row = idx >> 6, col = idx & 63;
            float4 a4 = *(const float4*)(ga + idx);
            float4 l4;
            l4.x = logf(1.0f - a4.x + 1e-10f);
            l4.y = logf(1.0f - a4.y + 1e-10f);
            l4.z = logf(1.0f - a4.z + 1e-10f);
            l4.w = logf(1.0f - a4.w + 1e-10f);
            *(float4*)(A + row * STRIDE + col) = a4;   // alpha
            *(float4*)(B + row * STRIDE + col) = l4;   // log terms
        }
    }
    __syncthreads();

    // ---------------- Phase 2: exclusive log-cumprod via WMMA ----------------
    // S[r,i] = sum_{k<i} log(1-a+1e-10)[r,k]   (B matrix = strict upper triangular ones)
    v8f acc[4];
    #pragma unroll
    for (int t = 0; t < 4; ++t) acc[t] = (v8f){0,0,0,0,0,0,0,0};

    for (int c = 0; c < 16; ++c) {               // K in chunks of 4
        int k0 = 4 * c + 2 * hh;
        v2f av;
        av.x = B[row16 * STRIDE + k0];
        av.y = B[row16 * STRIDE + k0 + 1];
        #pragma unroll
        for (int t = 0; t < 4; ++t) {
            int n = 16 * t + row16;
            v2f bv;
            bv.x = (k0     < n) ? 1.0f : 0.0f;
            bv.y = (k0 + 1 < n) ? 1.0f : 0.0f;
            acc[t] = __builtin_amdgcn_wmma_f32_16x16x4_f32(
                false, av, false, bv, (short)0, acc[t], false, false);
        }
    }
    __syncthreads();

    // ---------------- Phase 3: weights = alpha * exp(S); stage shifted pdf ----
    // pdf source = weights[:,1:63] + 1e-5 stored at k = i-1; k=62,63 zero pad.
    #pragma unroll
    for (int t = 0; t < 4; ++t) {
        int i = 16 * t + row16;                  // sample column (D layout N)
        #pragma unroll
        for (int j = 0; j < 8; ++j) {
            int r = j + 8 * hh;                  // ray row (D layout M)
            float w = A[r * STRIDE + i] * expf(acc[t][j]);
            if (i >= 1 && i <= 62) B[r * STRIDE + (i - 1)] = w + 1e-5f;
            if (i == 63) { B[r * STRIDE + 62] = 0.0f; B[r * STRIDE + 63] = 0.0f; }
        }
    }
    __syncthreads();

    // ---------------- Phase 4: inclusive cumsum via WMMA → raw CDF -----------
    // C[r,i] = sum_{k<=i} wplus[r,k]; cdf[0]=0, cdf[j]=C[r,j-1]; cdf[62]=total.
    v8f acc2[4];
    #pragma unroll
    for (int t = 0; t < 4; ++t) acc2[t] = (v8f){0,0,0,0,0,0,0,0};

    for (int c = 0; c < 16; ++c) {
        int k0 = 4 * c + 2 * hh;
        v2f av;
        av.x = B[row16 * STRIDE + k0];
        av.y = B[row16 * STRIDE + k0 + 1];
        #pragma unroll
        for (int t = 0; t < 4; ++t) {
            int n = 16 * t + row16;
            v2f bv;
            bv.x = (k0     <= n) ? 1.0f : 0.0f;
            bv.y = (k0 + 1 <= n) ? 1.0f : 0.0f;
            acc2[t] = __builtin_amdgcn_wmma_f32_16x16x4_f32(
                false, av, false, bv, (short)0, acc2[t], false, false);
        }
    }
    #pragma unroll
    for (int t = 0; t < 4; ++t) {
        int i = 16 * t + row16;
        #pragma unroll
        for (int j = 0; j < 8; ++j) {
            int r = j + 8 * hh;
            if (i <= 61) A[r * STRIDE + i + 1] = acc2[t][j];  // cdf[1..62]
            if (i == 63) A[r * STRIDE + 0] = 0.0f;            // cdf[0]
        }
    }
    __syncthreads();

    // ---------------- Phase 5: 128-sample inverse-CDF per ray ----------------
    // thread tid handles importance sample s = tid of each of the 64 rays.
    const int   s = tid;
    const float u = (float)s * (1.0f / 127.0f);

    for (int rr = 0; rr < RPB; ++rr) {
        const float* cdf = bufA[rr >> 4] + (rr & 15) * STRIDE;
        const int ray = blockRay0 + rr;

        float total = cdf[62];
        float inv_t = 1.0f / total;
        float ut    = u * total;                 // search raw CDF: identical result

        // searchsorted(cdf, u, side='right'): hi = first idx with cdf[hi] > ut
        int lo = 0, hi = 63;
        #pragma unroll
        for (int it = 0; it < 6; ++it) {
            int mid = (lo + hi) >> 1;            // probes 1..62
            bool le = (cdf[mid] <= ut);
            lo = le ? mid : lo;
            hi = le ? hi : mid;
        }
        int below = hi - 1;
        int above = (hi < 62) ? hi : 62;

        float c0 = cdf[below], c1 = cdf[above];
        float dn = (c1 - c0) * inv_t;
        dn = (dn < 1e-5f) ? 1.0f : dn;
        float tt = (u - c0 * inv_t) / dn;

        float nv = nearp[ray], fv = farp[ray];
        float span = fv - nv;
        float b0 = nv + span * ((float)below + 0.5f) * (1.0f / 63.0f);
        float b1 = nv + span * ((float)above + 0.5f) * (1.0f / 63.0f);
        float z  = b0 + tt * (b1 - b0);

        float ox = ray_o[ray * 3 + 0], oy = ray_o[ray * 3 + 1], oz = ray_o[ray * 3 + 2];
        float dx = ray_d[ray * 3 + 0], dy = ray_d[ray * 3 + 1], dz = ray_d[ray * 3 + 2];

        size_t pb = ((size_t)ray * NI + s) * 3;
        out[pb + 0] = ox + dx * z;
        out[pb + 1] = oy + dy * z;
        out[pb + 2] = oz + dz * z;
        out[(size_t)NRAYS * NI * 3 + (size_t)ray * NI + s] = z;
    }
}

extern "C" void kernel_launch(void* const* d_in, const int* in_sizes, int n_in,
                              void* d_out, int out_size, void* d_ws, size_t ws_size,
                              hipStream_t stream) {
    (void)in_sizes; (void)n_in; (void)out_size; (void)d_ws; (void)ws_size;
    const float* nearp = (const float*)d_in[0];
    const float* farp  = (const float*)d_in[1];
    const float* ro    = (const float*)d_in[2];
    const float* rd    = (const float*)d_in[3];
    const float* al    = (const float*)d_in[4];
    // d_in[5] = n_importance (==128, baked in as NI)
    float* out = (float*)d_out;

    dim3 grid(NRAYS / RPB);   // 4096 blocks x 128 threads (4 waves, 64 rays/block)
    nerf_resample_wmma<<<grid, 128, 0, stream>>>(nearp, farp, ro, rd, al, out);
}